// CustomRegionProposalNetwork_44032004718816
// MI455X (gfx1250) — compile-verified
//
#include <hip/hip_runtime.h>
#include <math.h>

// ---------------- problem constants (match reference) ----------------
#define Bn     8
#define An     400000
#define KPRE   2000
#define KPOST  1000
#define NBUCK  2048        // 11-bit radix buckets over flipped float keys
#define CAND   4096        // candidate pool per image (>= KPRE + radix-bucket ties)
#define NMS_T  0.7f
#define MIN_SZ 0.001f
#define XCLIP  4.135166556742356f   // log(1000/16)

// ---------------- helpers ----------------
__device__ __forceinline__ unsigned flip_key(float f) {
  // monotonic float->uint mapping: larger float => larger key
  unsigned u = __float_as_uint(f);
  return (u & 0x80000000u) ? ~u : (u | 0x80000000u);
}

// CDNA5 async global->LDS copy (VGLOBAL GLOBAL_LOAD_ASYNC_TO_LDS_B128).
// vdst VGPR carries the wave-relative LDS byte address; on gfx1250 the low
// 32 bits of a generic pointer to __shared__ data ARE that LDS offset
// (aperture check uses ADDR[63:32], LDS addressing uses ADDR[31:0]).
__device__ __forceinline__ void async_copy_b128(void* lds_dst, const void* gsrc) {
  unsigned           loff = (unsigned)(unsigned long long)lds_dst;
  unsigned long long ga   = (unsigned long long)gsrc;
  asm volatile("global_load_async_to_lds_b128 %0, %1, off"
               :: "v"(loff), "v"(ga) : "memory");
}

__device__ __forceinline__ void wait_async() {
#if __has_builtin(__builtin_amdgcn_s_wait_asynccnt)
  __builtin_amdgcn_s_wait_asynccnt(0);
#else
  asm volatile("s_wait_asynccnt 0x0" ::: "memory");
#endif
}

__device__ __forceinline__ void wait_tensor() {
#if __has_builtin(__builtin_amdgcn_s_wait_tensorcnt)
  __builtin_amdgcn_s_wait_tensorcnt(0);
#else
  asm volatile("s_wait_tensorcnt 0x0" ::: "memory");
#endif
}

// ---- Tensor Data Mover: 1-D tile DMA global->LDS via D# in SGPRs ----
// Descriptor per CDNA5 ISA:
//  group0: [1:0]=count=1, [63:32]=lds byte addr, [120:64]=global VA, [127:126]=type=2
//  group1: data_size=2 (4B), tensor_dim0=n, tensor_dim1=1, tile_dim0=n (<=65535),
//          tile_dim1=1, tensor_dim0_stride=n
typedef __attribute__((ext_vector_type(4))) unsigned tdm_g0_t;
typedef __attribute__((ext_vector_type(8))) int      tdm_g1_t;

__device__ __forceinline__ void tdm_load_1d(void* lds_dst, const void* gsrc,
                                            unsigned nelem4B) {
  unsigned long long ga = (unsigned long long)gsrc;
  tdm_g0_t g0;
  g0[0] = 1u;                                         // count=1 (user mode)
  g0[1] = (unsigned)(unsigned long long)lds_dst;      // LDS byte offset
  g0[2] = (unsigned)ga;                               // global_addr[31:0]
  g0[3] = ((unsigned)(ga >> 32) & 0x01FFFFFFu) | (2u << 30);  // [56:32] | type=2
  tdm_g1_t g1;
  g1[0] = (int)(2u << 16);                            // wg_mask=0, data_size=4B
  g1[1] = (int)((nelem4B & 0xFFFFu) << 16);           // tensor_dim0[15:0]
  g1[2] = (int)((nelem4B >> 16) | (1u << 16));        // dim0[31:16] | tensor_dim1=1
  g1[3] = (int)((nelem4B & 0xFFFFu) << 16);           // tile_dim0 (fits 16 bits)
  g1[4] = 1;                                          // tile_dim1=1, tile_dim2=0
  g1[5] = (int)nelem4B;                               // tensor_dim0_stride[31:0]
  g1[6] = 0;                                          // stride[47:32], dim1_stride lo
  g1[7] = 0;                                          // tensor_dim1_stride hi
  asm volatile("tensor_load_to_lds %0, %1" :: "s"(g0), "s"(g1) : "memory");
}

// ---------------- kernel 0: zero scratch ----------------
__global__ void zero_ws_kernel(unsigned* __restrict__ p, int n) {
  for (int i = blockIdx.x * blockDim.x + threadIdx.x; i < n;
       i += gridDim.x * blockDim.x)
    p[i] = 0u;
}

// ---------------- kernel 1: radix histogram (streaming pass 1) ----------------
__global__ void hist_kernel(const float* __restrict__ obj,
                            unsigned* __restrict__ hist) {
  __shared__ unsigned lh[NBUCK];
  const int img = blockIdx.y;
  for (int b = threadIdx.x; b < NBUCK; b += blockDim.x) lh[b] = 0u;
  __syncthreads();
  const float* o = obj + (size_t)img * An;
  const int stride = gridDim.x * blockDim.x;
  for (int i = blockIdx.x * blockDim.x + threadIdx.x; i < An; i += stride) {
    if (i + stride < An) __builtin_prefetch(o + i + stride, 0, 0);
    unsigned key = flip_key(o[i]);
    atomicAdd(&lh[key >> 21], 1u);
  }
  __syncthreads();
  unsigned* gh = hist + img * NBUCK;
  for (int b = threadIdx.x; b < NBUCK; b += blockDim.x)
    if (lh[b]) atomicAdd(&gh[b], lh[b]);
}

// ---------------- kernel 2: find threshold bucket per image ----------------
__global__ void thresh_kernel(const unsigned* __restrict__ hist,
                              unsigned* __restrict__ thr) {
  const int img = blockIdx.x;
  if (threadIdx.x == 0) {
    const unsigned* h = hist + img * NBUCK;
    unsigned run = 0, T = 0;
    for (int b = NBUCK - 1; b >= 0; --b) {
      run += h[b];
      if (run >= (unsigned)KPRE) { T = (unsigned)b; break; }
    }
    thr[img] = T;
  }
}

// ---------------- kernel 3: compact candidates (streaming pass 2, L2 hit) ----
__global__ void compact_kernel(const float* __restrict__ obj,
                               const unsigned* __restrict__ thr,
                               unsigned* __restrict__ cnt,
                               unsigned* __restrict__ ckey,
                               unsigned* __restrict__ cidx) {
  const int img = blockIdx.y;
  const unsigned T = thr[img];
  const float* o = obj + (size_t)img * An;
  const int stride = gridDim.x * blockDim.x;
  for (int i = blockIdx.x * blockDim.x + threadIdx.x; i < An; i += stride) {
    unsigned key = flip_key(o[i]);
    if ((key >> 21) >= T) {
      unsigned pos = atomicAdd(&cnt[img], 1u);
      if (pos < CAND) {
        ckey[img * CAND + pos] = key;
        cidx[img * CAND + pos] = (unsigned)i;
      }
    }
  }
}

// ---------------- kernel 4: bitonic sort (desc) + decode/clip top-2000 -------
__global__ void __launch_bounds__(1024)
sort_decode_kernel(const float* __restrict__ deltas,
                   const float* __restrict__ anchors,
                   const unsigned* __restrict__ ckey,
                   const unsigned* __restrict__ cidx,
                   const int* __restrict__ img_h, const int* __restrict__ img_w,
                   float* __restrict__ boxes, unsigned* __restrict__ keep) {
  __shared__ unsigned sk[CAND];
  __shared__ unsigned sv[CAND];
  const int img = blockIdx.x;
  const int tid = threadIdx.x;
  const int nt  = blockDim.x;
  for (int i = tid; i < CAND; i += nt) {
    sk[i] = ckey[img * CAND + i];
    sv[i] = cidx[img * CAND + i];
  }
  // bitonic sort, descending by key (padding keys are 0 => sink to the tail)
  for (unsigned k = 2; k <= CAND; k <<= 1) {
    for (unsigned j = k >> 1; j > 0; j >>= 1) {
      __syncthreads();
      for (unsigned idx = tid; idx < CAND; idx += nt) {
        unsigned p = idx ^ j;
        if (p > idx) {
          unsigned ki = sk[idx], kp = sk[p];
          bool desc = ((idx & k) == 0);
          if ((ki < kp) == desc) {
            sk[idx] = kp; sk[p] = ki;
            unsigned t = sv[idx]; sv[idx] = sv[p]; sv[p] = t;
          }
        }
      }
    }
  }
  __syncthreads();
  const float W = (float)img_w[0];
  const float H = (float)img_h[0];
  for (int r = tid; r < KPRE; r += nt) {
    const unsigned oi = sv[r];
    const float* a = anchors + ((size_t)img * An + oi) * 4;
    const float* d = deltas  + ((size_t)img * An + oi) * 4;
    float ax1 = a[0], ay1 = a[1], ax2 = a[2], ay2 = a[3];
    float w  = ax2 - ax1, h = ay2 - ay1;
    float cx = ax1 + 0.5f * w, cy = ay1 + 0.5f * h;
    float dx = d[0], dy = d[1];
    float dw = fminf(d[2], XCLIP), dh = fminf(d[3], XCLIP);
    float pcx = dx * w + cx, pcy = dy * h + cy;
    float pw  = expf(dw) * w, ph = expf(dh) * h;
    float x1 = pcx - 0.5f * pw, y1 = pcy - 0.5f * ph;
    float x2 = pcx + 0.5f * pw, y2 = pcy + 0.5f * ph;
    x1 = fminf(fmaxf(x1, 0.f), W); x2 = fminf(fmaxf(x2, 0.f), W);
    y1 = fminf(fmaxf(y1, 0.f), H); y2 = fminf(fmaxf(y2, 0.f), H);
    float* bo = boxes + ((size_t)img * 2048 + r) * 4;
    bo[0] = x1; bo[1] = y1; bo[2] = x2; bo[3] = y2;
    keep[img * 2048 + r] =
        ((x2 - x1) >= MIN_SZ && (y2 - y1) >= MIN_SZ) ? 1u : 0u;
  }
}

// ---------------- kernel 5: LDS-resident greedy NMS + masked output ----------
__global__ void __launch_bounds__(1024)
nms_out_kernel(const float* __restrict__ boxes,
               const unsigned* __restrict__ keepIn,
               float* __restrict__ outBoxes, float* __restrict__ outMask) {
  __shared__ __align__(16) float sb[KPRE * 4];  // 32 KB box tile
  __shared__ __align__(16) unsigned sk[KPRE];
  __shared__ unsigned srank[KPRE];
  __shared__ unsigned stotal;
  const int img = blockIdx.x;
  const int tid = threadIdx.x, nt = blockDim.x;
  const float* gb = boxes + (size_t)img * 2048 * 4;

  // Wave 0 issues one TDM DMA for the whole 32 KB box tile (TENSORcnt).
  if (tid < 32) {
    tdm_load_1d(sb, gb, (unsigned)(KPRE * 4));
    wait_tensor();
  }
  // Keep flags staged via per-lane async global->LDS B128 (ASYNCcnt).
  for (int i = tid; i < KPRE / 4; i += nt)
    async_copy_b128(&sk[4 * i], keepIn + (size_t)img * 2048 + 4 * i);
  wait_async();
  __syncthreads();

  // Sequential suppression (exactly the reference fori_loop semantics)
  for (int i = 0; i < KPRE; ++i) {
    if (sk[i]) {
      const float ax1 = sb[4 * i], ay1 = sb[4 * i + 1];
      const float ax2 = sb[4 * i + 2], ay2 = sb[4 * i + 3];
      const float areaA = (ax2 - ax1) * (ay2 - ay1);
      for (int j = i + 1 + tid; j < KPRE; j += nt) {
        if (sk[j]) {
          float bx1 = sb[4 * j], by1 = sb[4 * j + 1];
          float bx2 = sb[4 * j + 2], by2 = sb[4 * j + 3];
          float areaB = (bx2 - bx1) * (by2 - by1);
          float ix = fmaxf(fminf(ax2, bx2) - fmaxf(ax1, bx1), 0.f);
          float iy = fmaxf(fminf(ay2, by2) - fmaxf(ay1, by1), 0.f);
          float inter = ix * iy;
          float iou = inter / (areaA + areaB - inter + 1e-12f);
          if (iou > NMS_T) sk[j] = 0u;
        }
      }
    }
    __syncthreads();
  }

  if (tid == 0) {
    unsigned run = 0;
    for (int i = 0; i < KPRE; ++i) { srank[i] = run; run += sk[i] ? 1u : 0u; }
    stotal = run;
  }
  __syncthreads();

  float* ob = outBoxes + (size_t)img * KPOST * 4;
  float* om = outMask + (size_t)img * KPOST;
  const unsigned total = stotal;
  for (int p = tid; p < KPOST * 4; p += nt) ob[p] = 0.f;
  for (int p = tid; p < KPOST; p += nt)
    om[p] = ((unsigned)p < total) ? 1.f : 0.f;
  __syncthreads();
  for (int i = tid; i < KPRE; i += nt) {
    if (sk[i] && srank[i] < (unsigned)KPOST) {
      float* o = ob + 4 * srank[i];
      o[0] = sb[4 * i]; o[1] = sb[4 * i + 1];
      o[2] = sb[4 * i + 2]; o[3] = sb[4 * i + 3];
    }
  }
}

// ---------------- host entry ----------------
extern "C" void kernel_launch(void* const* d_in, const int* in_sizes, int n_in,
                              void* d_out, int out_size, void* d_ws,
                              size_t ws_size, hipStream_t stream) {
  const float* obj     = (const float*)d_in[0];  // (8, 400000)
  const float* deltas  = (const float*)d_in[1];  // (8, 400000, 4)
  const float* anchors = (const float*)d_in[2];  // (8, 400000, 4)
  const int*   ih      = (const int*)d_in[3];    // scalar
  const int*   iw      = (const int*)d_in[4];    // scalar
  (void)in_sizes; (void)n_in; (void)out_size; (void)ws_size;

  // workspace layout (u32 units); boxes/keep regions land 16B-aligned
  unsigned* ws    = (unsigned*)d_ws;
  unsigned* hist  = ws;                          // Bn*NBUCK
  unsigned* cnt   = hist + Bn * NBUCK;           // Bn
  unsigned* thr   = cnt + Bn;                    // Bn
  unsigned* ckey  = thr + Bn + 16;               // Bn*CAND (padded start)
  unsigned* cidx  = ckey + Bn * CAND;            // Bn*CAND
  float*    boxes = (float*)(cidx + Bn * CAND);  // Bn*2048*4 (16B-aligned)
  unsigned* keep  = (unsigned*)(boxes + Bn * 2048 * 4);  // Bn*2048 (16B-aligned)

  const int zeroN = (int)((ckey - ws) + 2 * Bn * CAND);  // hist..cidx
  zero_ws_kernel<<<256, 256, 0, stream>>>(ws, zeroN);

  dim3 sgrid(256, Bn);
  hist_kernel<<<sgrid, 256, 0, stream>>>(obj, hist);
  thresh_kernel<<<Bn, 32, 0, stream>>>(hist, thr);
  compact_kernel<<<sgrid, 256, 0, stream>>>(obj, thr, cnt, ckey, cidx);
  sort_decode_kernel<<<Bn, 1024, 0, stream>>>(deltas, anchors, ckey, cidx, ih,
                                              iw, boxes, keep);
  float* outBoxes = (float*)d_out;                 // (8,1000,4)
  float* outMask  = outBoxes + Bn * KPOST * 4;     // (8,1000)
  nms_out_kernel<<<Bn, 1024, 0, stream>>>(boxes, keep, outBoxes, outMask);
}